// NeighborList_78134045049163
// MI455X (gfx1250) — compile-verified
//
#include <hip/hip_runtime.h>
#include <hip/hip_bf16.h>
#include <stdint.h>

// Dense all-pairs PBC neighbor search (NNPOps getNeighborPairs dense mode).
// Output layout (floats): [P] i-or-neg1 | [P] j-or-neg1 | [P*3] deltas | [P] dists
#define CUTOFF_R 5.0f

// CDNA5 async copy: global memory -> LDS, 16 bytes per lane, tracked by ASYNCcnt.
// VDST = LDS byte address (low 32 bits of the flat shared pointer ARE the LDS
// offset per ISA 10.2 aperture mapping), VADDR = 64-bit global address, GV mode.
__device__ __forceinline__ void async_copy_b128(void* lds_dst, const void* gsrc) {
    uint32_t loff = (uint32_t)(uintptr_t)lds_dst;
    asm volatile("global_load_async_to_lds_b128 %0, %1, off"
                 :: "v"(loff), "v"(gsrc) : "memory");
}

__device__ __forceinline__ int row_start(int i, int n) {
    // first flat pair index of row i in triu(k=1) enumeration
    return i * n - ((i * (i + 1)) >> 1);
}

template <int PPT>
__global__ void __launch_bounds__(256)
neighbor_pairs_kernel(const float* __restrict__ pos,
                      const float* __restrict__ cell,
                      float* __restrict__ out,
                      int n, long long P) {
    extern __shared__ float s_pos[];   // n*3 floats (72 KB for n=6144)
    const int tid  = threadIdx.x;
    const int bdim = blockDim.x;

    // ---- Stage all positions into LDS with CDNA5 async-tensor path ----
    // Uniform trip count (exactly 18 full iterations for n=6144, 256 threads)
    // so no exec-mask loop idiom is needed; remainder loop is uniform too.
    const int nbytes = n * 3 * 4;
    const int step   = bdim * 16;
    const int full   = nbytes / step;
    int off = tid * 16;
    for (int it = 0; it < full; ++it, off += step) {
        async_copy_b128((char*)s_pos + off, (const char*)pos + off);
    }
    if (off < nbytes) {                 // tail (empty for n=6144)
        async_copy_b128((char*)s_pos + off, (const char*)pos + off);
    }
    asm volatile("s_wait_asynccnt 0x0" ::: "memory");
    __syncthreads();

    // ---- Cell + inverse (uniform across the grid; adjugate/det) ----
    const float c00 = cell[0], c01 = cell[1], c02 = cell[2];
    const float c10 = cell[3], c11 = cell[4], c12 = cell[5];
    const float c20 = cell[6], c21 = cell[7], c22 = cell[8];
    const float det = c00 * (c11 * c22 - c12 * c21)
                    - c01 * (c10 * c22 - c12 * c20)
                    + c02 * (c10 * c21 - c11 * c20);
    const float rd  = 1.0f / det;
    const float i00 = (c11 * c22 - c12 * c21) * rd;
    const float i01 = (c02 * c21 - c01 * c22) * rd;
    const float i02 = (c01 * c12 - c02 * c11) * rd;
    const float i10 = (c12 * c20 - c10 * c22) * rd;
    const float i11 = (c00 * c22 - c02 * c20) * rd;
    const float i12 = (c02 * c10 - c00 * c12) * rd;
    const float i20 = (c10 * c21 - c11 * c20) * rd;
    const float i21 = (c01 * c20 - c00 * c21) * rd;
    const float i22 = (c00 * c11 - c01 * c10) * rd;

    float* __restrict__ outI = out;
    float* __restrict__ outJ = out + P;
    float* __restrict__ outD = out + 2 * P;
    float* __restrict__ outR = out + 5 * P;

    const long long base = (long long)blockIdx.x * ((long long)bdim * PPT) + tid;
    const int t2 = 2 * n - 1;

    #pragma unroll
    for (int k = 0; k < PPT; ++k) {
        long long pl = base + (long long)k * bdim;
        if (pl >= P) break;
        int p = (int)pl;                       // P < 2^31, int32 math is exact

        // invert flat pair index -> (i, j): closed form, then BRANCHLESS
        // +/-1 corrections (sqrt abs error ~0.002 << 1, so two steps suffice;
        // ternaries lower to v_cmp + v_cndmask, no divergent branches).
        int   disc = t2 * t2 - 8 * p;          // fits int32 (max ~1.51e8)
        float sd   = __builtin_sqrtf((float)disc);
        int   i    = (int)(((float)t2 - sd) * 0.5f);
        i = i < 0 ? 0 : (i > n - 2 ? n - 2 : i);
        i -= (row_start(i, n) > p) ? 1 : 0;                          // safe: rs(0)=0<=p
        i += (i < n - 2 && row_start(i + 1, n) <= p) ? 1 : 0;
        i -= (row_start(i, n) > p) ? 1 : 0;                          // paranoia step
        i += (i < n - 2 && row_start(i + 1, n) <= p) ? 1 : 0;
        int j = i + 1 + (p - row_start(i, n));

        // pair displacement from LDS (stride-3 dwords: bank-conflict free)
        float dx = s_pos[3 * i + 0] - s_pos[3 * j + 0];
        float dy = s_pos[3 * i + 1] - s_pos[3 * j + 1];
        float dz = s_pos[3 * i + 2] - s_pos[3 * j + 2];

        // minimum image: frac = delta @ inv_cell; delta -= round(frac) @ cell
        float fx = dx * i00 + dy * i10 + dz * i20;
        float fy = dx * i01 + dy * i11 + dz * i21;
        float fz = dx * i02 + dy * i12 + dz * i22;
        float rx = __builtin_rintf(fx);        // RNE == jnp.round
        float ry = __builtin_rintf(fy);
        float rz = __builtin_rintf(fz);
        float wx = dx - (rx * c00 + ry * c10 + rz * c20);
        float wy = dy - (rx * c01 + ry * c11 + rz * c21);
        float wz = dz - (rx * c02 + ry * c12 + rz * c22);

        float dist = __builtin_sqrtf(wx * wx + wy * wy + wz * wz);
        bool  ok   = dist <= CUTOFF_R;

        // nontemporal streaming stores: 453 MB output > 192 MB L2, never re-read;
        // i/j/dist b32 stores are fully coalesced 128B lines per wave32; delta stores stride 12 B.
        __builtin_nontemporal_store(ok ? (float)i : -1.0f, outI + p);
        __builtin_nontemporal_store(ok ? (float)j : -1.0f, outJ + p);
        long long d3 = 3ll * p;
        __builtin_nontemporal_store(ok ? wx : 0.0f, outD + d3 + 0);
        __builtin_nontemporal_store(ok ? wy : 0.0f, outD + d3 + 1);
        __builtin_nontemporal_store(ok ? wz : 0.0f, outD + d3 + 2);
        __builtin_nontemporal_store(ok ? dist : 0.0f, outR + p);
    }
}

extern "C" void kernel_launch(void* const* d_in, const int* in_sizes, int n_in,
                              void* d_out, int out_size, void* d_ws, size_t ws_size,
                              hipStream_t stream) {
    const float* pos  = (const float*)d_in[0];   // [N,3] float32
    const float* cell = (const float*)d_in[1];   // [3,3] float32
    float*       out  = (float*)d_out;           // 6*P floats

    const int n = in_sizes[0] / 3;               // 6144
    const long long P = (long long)n * (n - 1) / 2;

    constexpr int BLOCK = 256;                   // 8 wave32 per block
    constexpr int PPT   = 16;                    // pairs per thread
    const long long pairsPerBlock = (long long)BLOCK * PPT;
    const int blocks = (int)((P + pairsPerBlock - 1) / pairsPerBlock);
    const size_t shmem = (size_t)n * 3 * sizeof(float);  // 73,728 B < 320 KB/WGP

    neighbor_pairs_kernel<PPT><<<blocks, BLOCK, shmem, stream>>>(pos, cell, out, n, P);
}